// GraphConvolution_44547400794649
// MI455X (gfx1250) — compile-verified
//
#include <hip/hip_runtime.h>
#include <hip/hip_bf16.h>

// GraphConvolution for MI455X (gfx1250, wave32).
//   support = X @ W          -- fp32 WMMA 16x16x4 (exact fp32 matrix pipe)
//   out     = bias (init)    -- broadcast
//   out[r] += v * support[c] -- one wave per edge, float4 gather + f32 atomics
//
// Sizes: N_NODES=100000 (=6250*16), N_EDGES=3200000, D_IN=D_OUT=128.

#define D_IN   128
#define D_OUT  128
#define N_NODES 100000
#define N_EDGES 3200000

typedef __attribute__((ext_vector_type(2))) float v2f;
typedef __attribute__((ext_vector_type(8))) float v8f;

// ---------------------------------------------------------------------------
// Kernel 1: support = X @ W via V_WMMA_F32_16X16X4_F32.
// Block = 256 threads = 8 waves. blockIdx.x = M-tile (16 rows).
// wave w (0..7) computes output cols [16w, 16w+16).
// K loop: 128 / 4 = 32 WMMA ops per wave.
//
// ISA VGPR layouts (CDNA5 ISA):
//   A 16x4 (32-bit): lanes 0-15 -> M=lane, {v0,v1}={K0,K1};
//                    lanes 16-31 -> M=lane-16, {v0,v1}={K2,K3}
//   B 4x16:          lanes 0-15 -> N=lane, {v0,v1}={K0,K1};
//                    lanes 16-31 -> N=lane-16, {v0,v1}={K2,K3}
//   C/D 16x16:       VGPR i: lanes 0-15 -> M=i, N=lane;
//                            lanes 16-31 -> M=i+8, N=lane-16
// ---------------------------------------------------------------------------
__global__ void gcn_gemm_wmma_f32(const float* __restrict__ X,
                                  const float* __restrict__ W,
                                  float* __restrict__ S) {
    const int mTile = blockIdx.x;                 // 0..6249
    const int wave  = threadIdx.x >> 5;           // 0..7  -> N tile
    const int lane  = threadIdx.x & 31;
    const int half  = lane >> 4;                  // 0 or 1
    const int l16   = lane & 15;

    const int aRow = mTile * 16 + l16;            // A-matrix row for this lane
    const int col  = wave * 16 + l16;             // B / D column for this lane

    const float* __restrict__ xrow = X + (size_t)aRow * D_IN;

    v8f acc = {};
#pragma unroll 8
    for (int kk = 0; kk < D_IN; kk += 4) {
        const int kA = kk + 2 * half;             // this lane's K pair base
        v2f a, b;
        a[0] = xrow[kA + 0];                      // contiguous pair -> b64 load
        a[1] = xrow[kA + 1];
        b[0] = W[(size_t)(kA + 0) * D_OUT + col]; // W row-major [K][N]
        b[1] = W[(size_t)(kA + 1) * D_OUT + col];
        acc = __builtin_amdgcn_wmma_f32_16x16x4_f32(
            /*neg_a=*/false, a, /*neg_b=*/false, b,
            /*c_mod=*/(short)0, acc, /*reuse_a=*/false, /*reuse_b=*/false);
    }

    // Store D: VGPR i holds row (mTile*16 + i + 8*half), column `col`.
    float* __restrict__ srow = S + ((size_t)(mTile * 16 + 8 * half)) * D_OUT + col;
#pragma unroll
    for (int i = 0; i < 8; ++i) {
        srow[(size_t)i * D_OUT] = acc[i];
    }
}

// ---------------------------------------------------------------------------
// Kernel 2: out[n][d] = bias[d]   (d_out is poisoned by the harness)
// ---------------------------------------------------------------------------
__global__ void gcn_bias_init(const float* __restrict__ bias,
                              float* __restrict__ out) {
    const int gid = blockIdx.x * blockDim.x + threadIdx.x;
    if (gid < N_NODES * D_OUT) {
        out[gid] = bias[gid & (D_OUT - 1)];
    }
}

// ---------------------------------------------------------------------------
// Kernel 3: scatter-accumulate. One wave (32 lanes) per edge; edge scalars are
// wave-uniform so the row/col/val loads become scalar-ish broadcasts. Each
// lane handles 4 consecutive floats: float4 gather of support row (hits L2:
// support is 51 MB < 192 MB) + 4 global_atomic_add_f32 into the dest row.
// ---------------------------------------------------------------------------
__global__ void gcn_edge_scatter(const int*   __restrict__ edge_row,
                                 const int*   __restrict__ edge_col,
                                 const float* __restrict__ edge_vals,
                                 const float* __restrict__ S,
                                 float* __restrict__ out) {
    const long long gid = (long long)blockIdx.x * blockDim.x + threadIdx.x;
    const long long e   = gid >> 5;
    const int lane      = (int)(gid & 31);
    if (e >= N_EDGES) return;

    const int   r = edge_row[e];
    const int   c = edge_col[e];
    const float v = edge_vals[e];

    const float4 sv = *reinterpret_cast<const float4*>(
        S + (size_t)c * D_OUT + lane * 4);        // global_load_b128
    float* o = out + (size_t)r * D_OUT + lane * 4;

    atomicAdd(o + 0, v * sv.x);                   // global_atomic_add_f32
    atomicAdd(o + 1, v * sv.y);
    atomicAdd(o + 2, v * sv.z);
    atomicAdd(o + 3, v * sv.w);
}

// ---------------------------------------------------------------------------
// Launch
// ---------------------------------------------------------------------------
extern "C" void kernel_launch(void* const* d_in, const int* in_sizes, int n_in,
                              void* d_out, int out_size, void* d_ws, size_t ws_size,
                              hipStream_t stream) {
    const float* x    = (const float*)d_in[0];   // [N_NODES, D_IN]
    const int*   er   = (const int*)  d_in[1];   // [N_EDGES] dest rows
    const int*   ec   = (const int*)  d_in[2];   // [N_EDGES] src cols
    const float* ev   = (const float*)d_in[3];   // [N_EDGES]
    const float* w    = (const float*)d_in[4];   // [D_IN, D_OUT]
    const float* bias = (const float*)d_in[5];   // [D_OUT]
    float*       out  = (float*)d_out;           // [N_NODES, D_OUT]
    float*       S    = (float*)d_ws;            // support scratch: 51.2 MB

    // 1) support = X @ W  (WMMA fp32)
    {
        dim3 grid(N_NODES / 16);                 // 6250 M-tiles, exact
        dim3 block(256);                         // 8 waves = 8 N-tiles
        gcn_gemm_wmma_f32<<<grid, block, 0, stream>>>(x, w, S);
    }

    // 2) out = bias
    {
        const int total = N_NODES * D_OUT;
        dim3 grid((total + 255) / 256);
        dim3 block(256);
        gcn_bias_init<<<grid, block, 0, stream>>>(bias, out);
    }

    // 3) out[r] += v * support[c]
    {
        const long long threads = (long long)N_EDGES * 32;
        dim3 grid((unsigned)((threads + 255) / 256));
        dim3 block(256);
        gcn_edge_scatter<<<grid, block, 0, stream>>>(er, ec, ev, S, out);
    }
}